// ConvKernel_6966436954611
// MI455X (gfx1250) — compile-verified
//
#include <hip/hip_runtime.h>
#include <hip/hip_bf16.h>
#include <math.h>

#define N_QUBITS 12
#define DIM 4096
#define BATCH 2048
#define HALF_DIM (DIM / 2)
#define SROW (16 * DIM / 2)   // v2f elements between 16-row blocks (256 KB)

typedef __attribute__((ext_vector_type(2))) float v2f;
typedef __attribute__((ext_vector_type(8))) float v8f;

#if defined(__has_builtin)
#if __has_builtin(__builtin_amdgcn_global_load_async_to_lds_b128) && \
    __has_builtin(__builtin_amdgcn_global_store_async_from_lds_b128)
#define USE_ASYNC_LDS 1
#endif
#endif

#ifdef USE_ASYNC_LDS
typedef int v4i __attribute__((vector_size(16)));
typedef __attribute__((address_space(1))) v4i* gptr_t;
typedef __attribute__((address_space(3))) v4i* lptr_t;
#endif

// ---------------------------------------------------------------------------
// prep: encoder columns col2[b,q,{a,b}] (complex), per-qubit R mats, final rot
// ---------------------------------------------------------------------------
__global__ void __launch_bounds__(256)
prep_kernel(const float* __restrict__ inputs,   // [BATCH, N_QUBITS]
            const float* __restrict__ weight,   // [3 + 2*N_QUBITS]
            float* __restrict__ col2,           // [BATCH*N_QUBITS*4]
            float* __restrict__ mats,           // [N_QUBITS*8]
            float* __restrict__ fin)            // [2]
{
    int idx = blockIdx.x * blockDim.x + threadIdx.x;
    if (idx < BATCH * N_QUBITS) {
        float x  = inputs[idx];
        float ry = 0.5f * x;
        float rz = 0.5f * x * x;
        float cy = cosf(ry), sy = sinf(ry);
        float cz = cosf(rz), sz = sinf(rz);
        float* o = col2 + (size_t)idx * 4;
        o[0] = cy * cz;  o[1] = -cy * sz;   // a = cos(ry) e^{-i rz}
        o[2] = sy * cz;  o[3] =  sy * sz;   // b = sin(ry) e^{+i rz}
    }
    if (idx < N_QUBITS) {
        float tx = 0.5f * weight[3 + idx];
        float tz = 0.5f * weight[3 + N_QUBITS + idx];
        float c = cosf(tx), s = sinf(tx);
        float cz = cosf(tz), sz = sinf(tz);
        float* m = mats + (size_t)idx * 8;
        m[0] =  c * cz;  m[1] = -c * sz;    // m00 =  c e^{-i tz}
        m[2] =  s * sz;  m[3] = -s * cz;    // m01 = -i s e^{+i tz}
        m[4] = -s * sz;  m[5] = -s * cz;    // m10 = -i s e^{-i tz}
        m[6] =  c * cz;  m[7] =  c * sz;    // m11 =  c e^{+i tz}
    }
    if (idx == N_QUBITS) {
        float r0 = 1.f, r1 = 0.f, r2 = 0.f, r3 = 1.f;
        for (int i = 0; i < 3; ++i) {
            float a = 0.5f * weight[i];
            float c = cosf(a), s = sinf(a);
            float n0 = c * r0 - s * r2, n1 = c * r1 - s * r3;
            float n2 = s * r0 + c * r2, n3 = s * r1 + c * r3;
            r0 = n0; r1 = n1; r2 = n2; r3 = n3;
        }
        fin[0] = r0; fin[1] = r1;
    }
}

// ---------------------------------------------------------------------------
// encode: v[b,s] = prod_q col2[b,q,bit_{11-q}(s)]  (qubit 0 = MSB)
// ---------------------------------------------------------------------------
__global__ void __launch_bounds__(256)
encode_kernel(const float* __restrict__ col2,
              float* __restrict__ vre, float* __restrict__ vim)
{
    size_t idx = (size_t)blockIdx.x * blockDim.x + threadIdx.x;
    int b = (int)(idx >> 12);
    int s = (int)(idx & (DIM - 1));
    const float* c = col2 + (size_t)b * N_QUBITS * 4;
    float re = 1.f, im = 0.f;
#pragma unroll
    for (int q = 0; q < N_QUBITS; ++q) {
        int bit = (s >> (N_QUBITS - 1 - q)) & 1;
        float cr = c[q * 4 + bit * 2 + 0];
        float ci = c[q * 4 + bit * 2 + 1];
        float nr = re * cr - im * ci;
        float ni = re * ci + im * cr;
        re = nr; im = ni;
    }
    vre[idx] = re;
    vim[idx] = im;
}

// ---------------------------------------------------------------------------
// GEMM: Y = X @ E^T for re+im planes. One wave per 64(M) x 32(N) tile:
// 16 f32 accumulators (128 VGPRs), 16 WMMAs per 10 b64 fragment loads.
// Software-pipelined with a UNIFORM scalar K counter (lane-dependent 'half'
// folded into the load index, not the induction variable) so the loop is
// scalar-controlled and exec stays trivially all-ones for the WMMAs.
// ---------------------------------------------------------------------------
__global__ void __launch_bounds__(32)
gemm_xEt(const float* __restrict__ Xre, const float* __restrict__ Xim,
         const float* __restrict__ E,
         float* __restrict__ Yre, float* __restrict__ Yim)
{
    const int lane = threadIdx.x & 31;
    const int half = lane >> 4;     // K sub-block: 0 -> K{0,1}, 1 -> K{2,3}
    const int l16  = lane & 15;
    const int mBase = blockIdx.y * 64;
    const int nBase = blockIdx.x * 32;

    const v2f* aRe = reinterpret_cast<const v2f*>(Xre + (size_t)(mBase + l16) * DIM);
    const v2f* aIm = reinterpret_cast<const v2f*>(Xim + (size_t)(mBase + l16) * DIM);
    const v2f* bE  = reinterpret_cast<const v2f*>(E   + (size_t)(nBase + l16) * DIM);

    __builtin_prefetch((const void*)bE, 0, 3);
    __builtin_prefetch((const void*)(bE + SROW), 0, 3);

    v8f accr[4][2] = {};
    v8f acci[4][2] = {};

    auto compute = [&](v2f b0, v2f b1, const v2f (&a0)[4], const v2f (&a1)[4]) {
#pragma unroll
        for (int s = 0; s < 4; ++s) {
            accr[s][0] = __builtin_amdgcn_wmma_f32_16x16x4_f32(
                false, a0[s], false, b0, (short)0, accr[s][0], false, false);
            accr[s][1] = __builtin_amdgcn_wmma_f32_16x16x4_f32(
                false, a0[s], false, b1, (short)0, accr[s][1], false, false);
            acci[s][0] = __builtin_amdgcn_wmma_f32_16x16x4_f32(
                false, a1[s], false, b0, (short)0, acci[s][0], false, false);
            acci[s][1] = __builtin_amdgcn_wmma_f32_16x16x4_f32(
                false, a1[s], false, b1, (short)0, acci[s][1], false, false);
        }
    };

    // prologue: fragments for kk = 0 (per-lane K index = half)
    v2f cb0 = bE[half];
    v2f cb1 = bE[half + SROW];
    v2f ca0[4], ca1[4];
#pragma unroll
    for (int s = 0; s < 4; ++s) {
        ca0[s] = aRe[half + s * SROW];
        ca1[s] = aIm[half + s * SROW];
    }

#pragma unroll 2
    for (int kk = 0; kk + 2 < DIM / 2; kk += 2) {      // uniform scalar counter
        const int kn = kk + 2 + half;                  // per-lane load index
        v2f nb0 = bE[kn];
        v2f nb1 = bE[kn + SROW];
        v2f na0[4], na1[4];
#pragma unroll
        for (int s = 0; s < 4; ++s) {
            na0[s] = aRe[kn + s * SROW];
            na1[s] = aIm[kn + s * SROW];
        }
        compute(cb0, cb1, ca0, ca1);
        cb0 = nb0; cb1 = nb1;
#pragma unroll
        for (int s = 0; s < 4; ++s) { ca0[s] = na0[s]; ca1[s] = na1[s]; }
    }
    compute(cb0, cb1, ca0, ca1);   // epilogue

    // C/D layout: VGPR r, lanes 0-15 -> M=r, lanes 16-31 -> M=8+r; N = lane&15
#pragma unroll
    for (int s = 0; s < 4; ++s) {
#pragma unroll
        for (int t = 0; t < 2; ++t) {
#pragma unroll
            for (int r = 0; r < 8; ++r) {
                int m = mBase + s * 16 + half * 8 + r;
                size_t o = (size_t)m * DIM + nBase + t * 16 + l16;
                Yre[o] = accr[s][t][r];
                Yim[o] = acci[s][t][r];
            }
        }
    }
}

// ---------------------------------------------------------------------------
// apply_R: u <- R u per batch row, 12 qubit-wise 2x2 complex butterflies.
// Row pair staged in 32 KB LDS via ASYNCcnt-tracked async LDS copies.
// ---------------------------------------------------------------------------
__global__ void __launch_bounds__(256)
apply_R_kernel(float* __restrict__ vre, float* __restrict__ vim,
               const float* __restrict__ mats)
{
    __shared__ float sre[DIM];
    __shared__ float sim[DIM];
    const int b = blockIdx.x;
    const int tid = threadIdx.x;
    float* rr = vre + (size_t)b * DIM;
    float* ri = vim + (size_t)b * DIM;

#ifdef USE_ASYNC_LDS
    for (int i = tid * 4; i < DIM; i += 256 * 4) {
        __builtin_amdgcn_global_load_async_to_lds_b128(
            (gptr_t)(rr + i), (lptr_t)(&sre[i]), 0, 0);
        __builtin_amdgcn_global_load_async_to_lds_b128(
            (gptr_t)(ri + i), (lptr_t)(&sim[i]), 0, 0);
    }
    asm volatile("s_wait_asynccnt 0x0" ::: "memory");
#else
    for (int i = tid; i < DIM; i += 256) { sre[i] = rr[i]; sim[i] = ri[i]; }
#endif
    __syncthreads();

    for (int q = 0; q < N_QUBITS; ++q) {
        const int p = N_QUBITS - 1 - q;
        const int stride = 1 << p;
        const float* m = mats + (size_t)q * 8;
        float m00r = m[0], m00i = m[1], m01r = m[2], m01i = m[3];
        float m10r = m[4], m10i = m[5], m11r = m[6], m11i = m[7];
        for (int t = tid; t < DIM / 2; t += 256) {
            int low = t & (stride - 1);
            int hi  = t >> p;
            int i0  = (hi << (p + 1)) | low;
            int i1  = i0 | stride;
            float x0r = sre[i0], x0i = sim[i0];
            float x1r = sre[i1], x1i = sim[i1];
            float y0r = m00r * x0r - m00i * x0i + m01r * x1r - m01i * x1i;
            float y0i = m00r * x0i + m00i * x0r + m01r * x1i + m01i * x1r;
            float y1r = m10r * x0r - m10i * x0i + m11r * x1r - m11i * x1i;
            float y1i = m10r * x0i + m10i * x0r + m11r * x1i + m11i * x1r;
            sre[i0] = y0r; sim[i0] = y0i;
            sre[i1] = y1r; sim[i1] = y1i;
        }
        __syncthreads();
    }

#ifdef USE_ASYNC_LDS
    for (int i = tid * 4; i < DIM; i += 256 * 4) {
        __builtin_amdgcn_global_store_async_from_lds_b128(
            (gptr_t)(rr + i), (lptr_t)(&sre[i]), 0, 0);
        __builtin_amdgcn_global_store_async_from_lds_b128(
            (gptr_t)(ri + i), (lptr_t)(&sim[i]), 0, 0);
    }
    asm volatile("s_wait_asynccnt 0x0" ::: "memory");
#else
    for (int i = tid; i < DIM; i += 256) { rr[i] = sre[i]; ri[i] = sim[i]; }
#endif
}

// ---------------------------------------------------------------------------
// final: out[b] = sum_{j<2048} |f00*z[b,j] + f01*z[b,j+2048]|^2
// ---------------------------------------------------------------------------
__global__ void __launch_bounds__(256)
final_kernel(const float* __restrict__ vre, const float* __restrict__ vim,
             const float* __restrict__ fin, float* __restrict__ out)
{
    __shared__ float red[8];
    const int b = blockIdx.x;
    const int tid = threadIdx.x;
    const float f00 = fin[0], f01 = fin[1];
    const float* rr = vre + (size_t)b * DIM;
    const float* ri = vim + (size_t)b * DIM;

    float acc = 0.f;
    for (int j = tid; j < HALF_DIM; j += 256) {
        float zr = f00 * rr[j] + f01 * rr[j + HALF_DIM];
        float zi = f00 * ri[j] + f01 * ri[j + HALF_DIM];
        acc += zr * zr + zi * zi;
    }
    for (int off = 16; off > 0; off >>= 1) acc += __shfl_down(acc, off, 32);
    if ((tid & 31) == 0) red[tid >> 5] = acc;
    __syncthreads();
    if (tid < 8) {
        float v = red[tid];
        for (int off = 4; off > 0; off >>= 1) v += __shfl_down(v, off, 8);
        if (tid == 0) out[b] = v;
    }
}

// ---------------------------------------------------------------------------
extern "C" void kernel_launch(void* const* d_in, const int* in_sizes, int n_in,
                              void* d_out, int out_size, void* d_ws, size_t ws_size,
                              hipStream_t stream)
{
    const float* inputs = (const float*)d_in[0];   // [2048,12]
    const float* weight = (const float*)d_in[1];   // [27]
    const float* E      = (const float*)d_in[2];   // [4096,4096]
    float* out = (float*)d_out;

    const size_t SZ = (size_t)BATCH * DIM;
    float* p0   = (float*)d_ws;                    // state re
    float* p1   = p0 + SZ;                         // state im
    float* p2   = p1 + SZ;                         // scratch re
    float* p3   = p2 + SZ;                         // scratch im
    float* col2 = p3 + SZ;                         // [BATCH*12*4]
    float* mats = col2 + (size_t)BATCH * N_QUBITS * 4;  // [12*8]
    float* fin  = mats + N_QUBITS * 8;             // [2]

    prep_kernel<<<(BATCH * N_QUBITS + 255) / 256, 256, 0, stream>>>(
        inputs, weight, col2, mats, fin);

    encode_kernel<<<(unsigned)(SZ / 256), 256, 0, stream>>>(col2, p0, p1);

    dim3 gemm_grid(DIM / 32, BATCH / 64);          // 128 x 32 tiles
    gemm_xEt<<<gemm_grid, 32, 0, stream>>>(p0, p1, E, p2, p3);   // v @ E^T

    apply_R_kernel<<<BATCH, 256, 0, stream>>>(p2, p3, mats);     // R qubit-wise

    gemm_xEt<<<gemm_grid, 32, 0, stream>>>(p2, p3, E, p0, p1);   // @ E^T again

    final_kernel<<<BATCH, 256, 0, stream>>>(p0, p1, fin, out);
}